// GCNEncoder_11570641895553
// MI455X (gfx1250) — compile-verified
//
#include <hip/hip_runtime.h>
#include <hip/hip_bf16.h>

#define NDIM 128              // IN_DIM == OUT_DIM == 128
#define NEG_SLOPE 0.01f

typedef __attribute__((ext_vector_type(2))) float v2f;
typedef __attribute__((ext_vector_type(8))) float v8f;

// ---------------------------------------------------------------------------
// Transpose 128x128 W (row-major [out][in]) -> Wt (row-major [in][out]) so the
// WMMA B-operand loads are lane-coalesced. 64 KB, stays hot in L2.
// ---------------------------------------------------------------------------
__global__ void transpose128(const float* __restrict__ W, float* __restrict__ Wt) {
    int i = blockIdx.x * blockDim.x + threadIdx.x;   // 0 .. 16383
    if (i < NDIM * NDIM) {
        int r = i >> 7, c = i & 127;                 // W[r][c]
        Wt[c * NDIM + r] = W[i];
    }
}

// ---------------------------------------------------------------------------
// Degree pipeline: dis <- 1.0 ; dis[dst[e]] += 1 ; dis <- rsqrt(dis)
// ---------------------------------------------------------------------------
__global__ void fill_ones(float* __restrict__ p, int n) {
    int i = blockIdx.x * blockDim.x + threadIdx.x;
    if (i < n) p[i] = 1.0f;
}

__global__ void count_deg(const int* __restrict__ dst, float* __restrict__ deg, int e) {
    int i = blockIdx.x * blockDim.x + threadIdx.x;
    if (i < e) atomicAdd(&deg[dst[i]], 1.0f);
}

__global__ void rsqrt_inplace(float* __restrict__ p, int n) {
    int i = blockIdx.x * blockDim.x + threadIdx.x;
    if (i < n) p[i] = rsqrtf(p[i]);
}

__global__ void zero_f32(float* __restrict__ p, int n) {
    int i = blockIdx.x * blockDim.x + threadIdx.x;
    if (i < n) p[i] = 0.0f;
}

// ---------------------------------------------------------------------------
// H[N][128] = X[N][128] @ Wt[128][128]  via V_WMMA_F32_16X16X4_F32.
// Block = 256 threads = 8 waves; each wave owns one 16x16 output tile, the 8
// waves cover all 128 output columns of a 16-row slab. X slab staged in LDS
// (padded stride 132 -> conflict-free ds_load_b64 A-fragment reads).
// ---------------------------------------------------------------------------
__global__ void __launch_bounds__(256)
gemm_wmma_f32(const float* __restrict__ X, const float* __restrict__ Wt,
              float* __restrict__ H, int nrows) {
    __shared__ float Xs[16][132];                    // 16 rows x 128 (+4 pad)

    const int tid     = threadIdx.x;
    const int rowBase = blockIdx.x * 16;

    // Cooperative load of the 16x128 X slab (8 float loads per thread).
    for (int i = tid; i < 16 * NDIM; i += 256) {
        int r = i >> 7, c = i & 127;
        int gr = rowBase + r;
        Xs[r][c] = (gr < nrows) ? X[gr * NDIM + c] : 0.0f;
    }
    __syncthreads();

    const int wave    = tid >> 5;                    // 0..7  -> column tile
    const int lane    = tid & 31;
    const int colBase = wave * 16;
    const int l16     = lane & 15;
    const int khalf   = (lane >> 4) << 1;            // 0 for lanes 0-15, 2 for 16-31

    v8f acc = {};
    #pragma unroll
    for (int k = 0; k < NDIM; k += 4) {
        v2f a, b;
        // A (16x4 f32): lane l16 = row; VGPR0/1 = K {k+khalf, k+khalf+1}
        a.x = Xs[l16][k + khalf];
        a.y = Xs[l16][k + khalf + 1];
        // B (4x16 f32): lane l16 = col; VGPR0/1 = K {k+khalf, k+khalf+1}
        b.x = Wt[(k + khalf)     * NDIM + colBase + l16];
        b.y = Wt[(k + khalf + 1) * NDIM + colBase + l16];
        acc = __builtin_amdgcn_wmma_f32_16x16x4_f32(
            /*neg_a=*/false, a, /*neg_b=*/false, b,
            /*c_mod=*/(short)0, acc, /*reuse_a=*/false, /*reuse_b=*/false);
    }

    // C/D layout: VGPR v -> row (v + 8*(lane>=16)), col = lane%16.
    const int row0 = rowBase + ((lane >> 4) << 3);
    const int col  = colBase + l16;
    #pragma unroll
    for (int v = 0; v < 8; ++v) {
        int r = row0 + v;
        if (r < nrows) H[r * NDIM + col] = acc[v];
    }
}

// ---------------------------------------------------------------------------
// Edge scatter: 32 threads per edge, float4 gather of h[src], 4 f32 global
// atomics into agg[dst]. Both h and agg are L2-resident (25.6 MB each).
// ---------------------------------------------------------------------------
__global__ void __launch_bounds__(256)
edge_scatter(const float* __restrict__ h, const float* __restrict__ dis,
             const int* __restrict__ src, const int* __restrict__ dst,
             float* __restrict__ agg, int e) {
    int gid  = blockIdx.x * blockDim.x + threadIdx.x;
    int edge = gid >> 5;                             // 32 lanes per edge
    if (edge >= e) return;
    int part = gid & 31;                             // 4 floats per lane

    int s = src[edge];
    int d = dst[edge];
    float norm = dis[s] * dis[d];

    const float4 hv = ((const float4*)(h + (size_t)s * NDIM))[part];
    float* out = agg + (size_t)d * NDIM + part * 4;
    atomicAdd(out + 0, hv.x * norm);
    atomicAdd(out + 1, hv.y * norm);
    atomicAdd(out + 2, hv.z * norm);
    atomicAdd(out + 3, hv.w * norm);
}

// ---------------------------------------------------------------------------
// Fused epilogue: agg += h * dis^2 (self-loop) + bias ; optional leaky-ReLU.
// In-place on agg.
// ---------------------------------------------------------------------------
__global__ void __launch_bounds__(256)
post_epilogue(float* __restrict__ agg, const float* __restrict__ h,
              const float* __restrict__ dis, const float* __restrict__ bias,
              int n, int do_lrelu) {
    int i = blockIdx.x * blockDim.x + threadIdx.x;
    if (i >= n * NDIM) return;
    int row = i >> 7, c = i & 127;
    float ds = dis[row];
    float v = agg[i] + h[i] * (ds * ds) + bias[c];
    if (do_lrelu) v = (v > 0.0f) ? v : NEG_SLOPE * v;
    agg[i] = v;
}

// ---------------------------------------------------------------------------
// kernel_launch
// Inputs: 0:x[N*128] 1:edge_index[2*E] 2:batch[N] 3:W1[128*128] 4:b1[128]
//         5:W2[128*128] 6:b2[128]
// ---------------------------------------------------------------------------
extern "C" void kernel_launch(void* const* d_in, const int* in_sizes, int n_in,
                              void* d_out, int out_size, void* d_ws, size_t ws_size,
                              hipStream_t stream) {
    const float* x    = (const float*)d_in[0];
    const int*   ei   = (const int*)d_in[1];
    const float* W1   = (const float*)d_in[3];
    const float* b1   = (const float*)d_in[4];
    const float* W2   = (const float*)d_in[5];
    const float* b2   = (const float*)d_in[6];

    const int N = in_sizes[0] / NDIM;       // 50000
    const int E = in_sizes[1] / 2;          // 800000
    const int* src = ei;
    const int* dst = ei + E;

    // Workspace carve-up (all offsets 16B-aligned).
    float* Wt1  = (float*)d_ws;                 // 16384
    float* Wt2  = Wt1 + NDIM * NDIM;            // 16384
    float* dis  = Wt2 + NDIM * NDIM;            // N
    float* bufA = dis + ((N + 3) & ~3);         // N*128  (h)
    float* bufB = bufA + (size_t)N * NDIM;      // N*128  (agg / layer-1 out)
    float* outp = (float*)d_out;

    const int T = 256;
    dim3 blk(T);
    dim3 gW((NDIM * NDIM + T - 1) / T);         // 64
    dim3 gN((N + T - 1) / T);
    dim3 gE((E + T - 1) / T);
    dim3 gNF((N * NDIM + T - 1) / T);
    dim3 gScat(((E * 32) + T - 1) / T);
    dim3 gGemm((N + 15) / 16);

    // 0) weight transposes + degree / rsqrt (shared by both layers)
    transpose128<<<gW, blk, 0, stream>>>(W1, Wt1);
    transpose128<<<gW, blk, 0, stream>>>(W2, Wt2);
    fill_ones<<<gN, blk, 0, stream>>>(dis, N);
    count_deg<<<gE, blk, 0, stream>>>(dst, dis, E);
    rsqrt_inplace<<<gN, blk, 0, stream>>>(dis, N);

    // 1) layer 1: h1 = x @ W1^T ; agg = scatter ; lrelu(agg + h1*dis^2 + b1)
    gemm_wmma_f32<<<gGemm, blk, 0, stream>>>(x, Wt1, bufA, N);
    zero_f32<<<gNF, blk, 0, stream>>>(bufB, N * NDIM);
    edge_scatter<<<gScat, blk, 0, stream>>>(bufA, dis, src, dst, bufB, E);
    post_epilogue<<<gNF, blk, 0, stream>>>(bufB, bufA, dis, b1, N, /*lrelu=*/1);

    // 2) layer 2: h2 = h1out @ W2^T ; scatter into d_out ; final epilogue
    gemm_wmma_f32<<<gGemm, blk, 0, stream>>>(bufB, Wt2, bufA, N);
    zero_f32<<<gNF, blk, 0, stream>>>(outp, N * NDIM);
    edge_scatter<<<gScat, blk, 0, stream>>>(bufA, dis, src, dst, outp, E);
    post_epilogue<<<gNF, blk, 0, stream>>>(outp, bufA, dis, b2, N, /*lrelu=*/0);
}